// GCNLinkPredictor_62371515072932
// MI455X (gfx1250) — compile-verified
//
#include <hip/hip_runtime.h>
#include <hip/hip_bf16.h>
#include <math.h>

typedef float v2f __attribute__((ext_vector_type(2)));
typedef float v8f __attribute__((ext_vector_type(8)));

#define HID 64
#define IN_C 128

// ---------------- degree / normalization ----------------

__global__ void k_init_deg(float* deg, int N) {
  int i = blockIdx.x * blockDim.x + threadIdx.x;
  if (i < N) deg[i] = 1.0f;  // self-loop contributes 1 to every node's degree
}

__global__ void k_deg_accum(const long long* __restrict__ ei, float* deg, int E) {
  int e = blockIdx.x * blockDim.x + threadIdx.x;
  if (e < E) {
    long long d = ei[(size_t)E + e];  // dst row of [2,E]
    atomicAdd(&deg[d], 1.0f);
  }
}

__global__ void k_dinv(float* deg, int N) {
  int i = blockIdx.x * blockDim.x + threadIdx.x;
  if (i < N) deg[i] = rsqrtf(deg[i]);  // deg >= 1 always
}

__global__ void k_norm(const long long* __restrict__ ei, const float* __restrict__ dinv,
                       float* __restrict__ nrm, int E) {
  int e = blockIdx.x * blockDim.x + threadIdx.x;
  if (e < E) {
    long long s = ei[e];
    long long d = ei[(size_t)E + e];
    nrm[e] = dinv[s] * dinv[d];
  }
}

// ---------------- dense GEMM via f32 WMMA ----------------
// H[M,64] = X[M,K] @ W[K,64]; one wave computes a 16x16 tile with
// V_WMMA_F32_16X16X4_F32, K stepped by 4. EXEC is all-1s throughout the loop.

__global__ void k_gemm_wmma(const float* __restrict__ X, const float* __restrict__ W,
                            float* __restrict__ H, int M, int K) {
  int gtid = blockIdx.x * blockDim.x + threadIdx.x;
  int wave = gtid >> 5;
  int lane = threadIdx.x & 31;
  int mTile = wave >> 2;            // 4 column tiles (N = 64)
  int nTile = wave & 3;
  int m0 = mTile * 16;
  if (m0 >= M) return;              // uniform per wave
  int n0 = nTile * 16;
  int half = lane >> 4;             // 0: lanes 0-15, 1: lanes 16-31
  int l16  = lane & 15;

  // clamp row for safety (M is a multiple of 16 here, so normally exact)
  int mrow = m0 + l16; if (mrow > M - 1) mrow = M - 1;
  const float* xrow = X + (size_t)mrow * K;

  v8f acc = {};
  for (int kk = 0; kk < K; kk += 4) {
    int k = kk + 2 * half;
    v2f a, b;
    a[0] = xrow[k];
    a[1] = xrow[k + 1];
    b[0] = W[(size_t)k * HID + n0 + l16];
    b[1] = W[(size_t)(k + 1) * HID + n0 + l16];
    // D = A(16x4 f32) x B(4x16 f32) + C  -> v_wmma_f32_16x16x4_f32
    acc = __builtin_amdgcn_wmma_f32_16x16x4_f32(
        /*neg_a=*/false, a, /*neg_b=*/false, b,
        /*c_mod=*/(short)0, acc, /*reuse_a=*/false, /*reuse_b=*/false);
  }

  // D layout: VGPR r holds row m0+r (lanes 0-15) / m0+r+8 (lanes 16-31), col n0+l16
#pragma unroll
  for (int r = 0; r < 8; ++r) {
    int m = m0 + r + 8 * half;
    if (m < M) H[(size_t)m * HID + n0 + l16] = acc[r];
  }
}

// ---------------- scatter aggregation ----------------
// 16 threads per edge, float4 per thread: gather h[src], scale by norm, atomic-add to out[dst].

__global__ void k_zero(float* __restrict__ p, long long n) {
  long long t = (long long)blockIdx.x * blockDim.x + threadIdx.x;
  if (t < n) p[t] = 0.0f;
}

__global__ void k_agg(const float* __restrict__ Hin, const long long* __restrict__ ei,
                      const float* __restrict__ nrm, float* __restrict__ Hout, int E) {
  long long t = (long long)blockIdx.x * blockDim.x + threadIdx.x;
  int e = (int)(t >> 4);
  if (e >= E) return;
  int f = ((int)t & 15) * 4;
  long long s = ei[e];
  long long d = ei[(size_t)E + e];
  float w = nrm[e];
  float4 v = *(const float4*)(Hin + (size_t)s * HID + f);
  float* o = Hout + (size_t)d * HID + f;
  atomicAdd(o + 0, v.x * w);
  atomicAdd(o + 1, v.y * w);
  atomicAdd(o + 2, v.z * w);
  atomicAdd(o + 3, v.w * w);
}

// self-loop term + bias + optional relu:  out[i] += h_lin[i]*dinv[i]^2 + b; relu
__global__ void k_finish(const float* __restrict__ Hlin, const float* __restrict__ dinv,
                         const float* __restrict__ bias, float* __restrict__ Hout,
                         long long n64, int relu) {
  long long t = (long long)blockIdx.x * blockDim.x + threadIdx.x;
  if (t >= n64) return;
  int node = (int)(t >> 6);
  int f = (int)(t & 63);
  float di = dinv[node];
  float v = Hout[t] + Hlin[t] * di * di + bias[f];
  Hout[t] = relu ? fmaxf(v, 0.0f) : v;
}

// ---------------- edge scoring ----------------
// 16 lanes per label edge: float4 partial dot, xor-shuffle reduce within 16 lanes.

__global__ void k_score(const float* __restrict__ H, const long long* __restrict__ eli,
                        float* __restrict__ out, int EL) {
  long long t = (long long)blockIdx.x * blockDim.x + threadIdx.x;
  int e = (int)(t >> 4);
  if (e >= EL) return;
  int f = ((int)t & 15) * 4;
  long long s = eli[e];
  long long d = eli[(size_t)EL + e];
  float4 a = *(const float4*)(H + (size_t)s * HID + f);
  float4 b = *(const float4*)(H + (size_t)d * HID + f);
  float p = a.x * b.x + a.y * b.y + a.z * b.z + a.w * b.w;
#pragma unroll
  for (int off = 8; off >= 1; off >>= 1) p += __shfl_xor(p, off, 16);
  if (((int)t & 15) == 0) out[e] = 1.0f / (1.0f + expf(-p));
}

// ---------------- driver ----------------

extern "C" void kernel_launch(void* const* d_in, const int* in_sizes, int n_in,
                              void* d_out, int out_size, void* d_ws, size_t ws_size,
                              hipStream_t stream) {
  const float*     x   = (const float*)d_in[0];
  const float*     W1  = (const float*)d_in[1];
  const float*     b1  = (const float*)d_in[2];
  const float*     W2  = (const float*)d_in[3];
  const float*     b2  = (const float*)d_in[4];
  const float*     W3  = (const float*)d_in[5];
  const float*     b3  = (const float*)d_in[6];
  const long long* ei  = (const long long*)d_in[7];  // int64 [2,E]
  const long long* eli = (const long long*)d_in[8];  // int64 [2,EL]
  float* scores = (float*)d_out;

  int N  = in_sizes[0] / IN_C;
  int E  = in_sizes[7] / 2;
  int EL = in_sizes[8] / 2;

  // carve workspace: dinv | norm | A | B
  char* base = (char*)d_ws;
  size_t off = 0;
  auto carve = [&](size_t bytes) -> void* {
    void* p = base + off;
    off = (off + bytes + 255) & ~(size_t)255;
    return p;
  };
  float* dinv = (float*)carve((size_t)N * sizeof(float));
  float* nrm  = (float*)carve((size_t)E * sizeof(float));
  float* A    = (float*)carve((size_t)N * HID * sizeof(float));
  float* B    = (float*)carve((size_t)N * HID * sizeof(float));
  (void)ws_size; (void)n_in; (void)out_size;

  const int T = 256;
  int nBlocks   = (N + T - 1) / T;
  int eBlocks   = (E + T - 1) / T;
  long long n64 = (long long)N * HID;
  int n64Blocks = (int)((n64 + T - 1) / T);
  long long aggThreads = (long long)E * 16;
  int aggBlocks = (int)((aggThreads + T - 1) / T);
  long long scThreads = (long long)EL * 16;
  int scBlocks = (int)((scThreads + T - 1) / T);
  int mTiles = (N + 15) / 16;
  int gemmBlocks = (mTiles * 4 * 32 + T - 1) / T;

  // --- normalization coefficients (shared by all 3 layers) ---
  k_init_deg <<<nBlocks, T, 0, stream>>>(dinv, N);
  k_deg_accum<<<eBlocks, T, 0, stream>>>(ei, dinv, E);
  k_dinv     <<<nBlocks, T, 0, stream>>>(dinv, N);
  k_norm     <<<eBlocks, T, 0, stream>>>(ei, dinv, nrm, E);

  // --- layer 1: A = x@W1 ; B = scatter(A) + self + b1 ; relu ---
  k_gemm_wmma<<<gemmBlocks, T, 0, stream>>>(x, W1, A, N, IN_C);
  k_zero     <<<n64Blocks, T, 0, stream>>>(B, n64);
  k_agg      <<<aggBlocks, T, 0, stream>>>(A, ei, nrm, B, E);
  k_finish   <<<n64Blocks, T, 0, stream>>>(A, dinv, b1, B, n64, 1);

  // --- layer 2: A = B@W2 ; B = scatter(A) + self + b2 ; relu ---
  k_gemm_wmma<<<gemmBlocks, T, 0, stream>>>(B, W2, A, N, HID);
  k_zero     <<<n64Blocks, T, 0, stream>>>(B, n64);
  k_agg      <<<aggBlocks, T, 0, stream>>>(A, ei, nrm, B, E);
  k_finish   <<<n64Blocks, T, 0, stream>>>(A, dinv, b2, B, n64, 1);

  // --- layer 3: A = B@W3 ; B = scatter(A) + self + b3 (no relu) ---
  k_gemm_wmma<<<gemmBlocks, T, 0, stream>>>(B, W3, A, N, HID);
  k_zero     <<<n64Blocks, T, 0, stream>>>(B, n64);
  k_agg      <<<aggBlocks, T, 0, stream>>>(A, ei, nrm, B, E);
  k_finish   <<<n64Blocks, T, 0, stream>>>(A, dinv, b3, B, n64, 0);

  // --- link scores ---
  k_score<<<scBlocks, T, 0, stream>>>(B, eli, scores, EL);
}